// GCN_72000831750590
// MI455X (gfx1250) — compile-verified
//
#include <hip/hip_runtime.h>

typedef float v2f __attribute__((ext_vector_type(2)));
typedef float v8f __attribute__((ext_vector_type(8)));

#define NFEAT 128
#define LDSW_STRIDE 132   // 128 + 4 floats pad: avoids 64-bank conflicts on column reads

// ---------------------------------------------------------------- init: zero out, deg=1 (self loop)
__global__ void gcn_init(float* __restrict__ out, unsigned* __restrict__ deg,
                         int n_nodes, long long total) {
  long long i = (long long)blockIdx.x * blockDim.x + threadIdx.x;
  if (i < total)   out[i] = 0.0f;
  if (i < n_nodes) deg[i] = 1u;
}

// ---------------------------------------------------------------- degree histogram over source ids
__global__ void gcn_degree(const long long* __restrict__ row,
                           unsigned* __restrict__ deg, int nE) {
  int e = blockIdx.x * blockDim.x + threadIdx.x;
  if (e < nE) atomicAdd(&deg[(int)row[e]], 1u);
}

// ---------------------------------------------------------------- deg -> rsqrt(deg), in place
__global__ void gcn_rsqrt(unsigned* __restrict__ degdis, int n_nodes) {
  int i = blockIdx.x * blockDim.x + threadIdx.x;
  if (i < n_nodes) {
    float d = (float)degdis[i];
    ((float*)degdis)[i] = rsqrtf(d);
  }
}

// ---------------------------------------------------------------- X = F @ W^T + b via f32 WMMA
// one wave per 16-row strip; 8 column tiles of 16; K loop of 32 x V_WMMA_F32_16X16X4_F32
__global__ __launch_bounds__(256) void gcn_gemm(const float* __restrict__ F,
                                                const float* __restrict__ W,
                                                const float* __restrict__ bias,
                                                float* __restrict__ X,
                                                int n_nodes, int n_strips) {
  __shared__ float ldsW[NFEAT * LDSW_STRIDE];   // 66 KB of the 320 KB WGP LDS

  int tid = threadIdx.x;
  for (int i = tid; i < NFEAT * NFEAT; i += 256) {
    ldsW[(i >> 7) * LDSW_STRIDE + (i & 127)] = W[i];
  }
  __syncthreads();

  int strip = blockIdx.x * 8 + (tid >> 5);      // wave-uniform
  if (strip >= n_strips) return;                // whole wave exits together

  int lane = tid & 31;
  int lo = lane & 15;                           // column-within-tile / row-within-strip
  int hi = lane >> 4;                           // K-pair selector (A/B), row-half selector (C/D)
  int mbase = strip * 16;

  // clamped A row pointer so EXEC stays all-ones even on a ragged tail
  int mrow = mbase + lo;
  if (mrow >= n_nodes) mrow = n_nodes - 1;
  const float* arow = F + (size_t)mrow * NFEAT + hi * 2;

  v8f acc[8];
#pragma unroll
  for (int nt = 0; nt < 8; ++nt) {
    float bv = bias[nt * 16 + lo];              // C[m][n] starts at b[n] for every m
    acc[nt] = (v8f){bv, bv, bv, bv, bv, bv, bv, bv};
  }

  for (int k = 0; k < NFEAT; k += 4) {
    v2f a = *(const v2f*)(arow + k);            // A: lanes 0-15 K=k..k+1, lanes 16-31 K=k+2..k+3
#pragma unroll
    for (int nt = 0; nt < 8; ++nt) {
      // B[k][n] = W[n][k]; contiguous K-pair from LDS row n
      const float* wp = &ldsW[(nt * 16 + lo) * LDSW_STRIDE + k + hi * 2];
      v2f bf = *(const v2f*)wp;
      acc[nt] = __builtin_amdgcn_wmma_f32_16x16x4_f32(
          /*neg_a=*/false, a, /*neg_b=*/false, bf,
          /*c_mod=*/(short)0, acc[nt], /*reuse_a=*/false, /*reuse_b=*/false);
    }
  }

  // D layout: lanes 0-15 rows mbase+0..7, lanes 16-31 rows mbase+8..15 (VGPR v = row offset)
#pragma unroll
  for (int nt = 0; nt < 8; ++nt) {
#pragma unroll
    for (int v = 0; v < 8; ++v) {
      int m = mbase + hi * 8 + v;
      if (m < n_nodes)
        X[(size_t)m * NFEAT + nt * 16 + lo] = acc[nt][v];
    }
  }
}

// ---------------------------------------------------------------- edge scatter: one wave per edge
__global__ void gcn_scatter(const long long* __restrict__ row,
                            const long long* __restrict__ col,
                            const float* __restrict__ X,
                            const float* __restrict__ dis,
                            float* __restrict__ out,
                            long long nE, int n_nodes) {
  long long wid = ((long long)blockIdx.x * blockDim.x + threadIdx.x) >> 5;  // wave-uniform
  long long tot = nE + (long long)n_nodes;
  if (wid >= tot) return;
  int lane = threadIdx.x & 31;

  int r, c;
  float nrm;
  if (wid < nE) {
    r = (int)row[wid];
    c = (int)col[wid];
    nrm = dis[r] * dis[c];
  } else {                      // self loop: row = col = node, norm = 1/deg
    r = c = (int)(wid - nE);
    float d = dis[r];
    nrm = d * d;
  }

  float4 v = *((const float4*)(X + (size_t)r * NFEAT) + lane);  // 128 floats / wave, L2-resident
  float* o = out + (size_t)c * NFEAT + lane * 4;
  unsafeAtomicAdd(o + 0, v.x * nrm);   // global_atomic_add_f32, no return
  unsafeAtomicAdd(o + 1, v.y * nrm);
  unsafeAtomicAdd(o + 2, v.z * nrm);
  unsafeAtomicAdd(o + 3, v.w * nrm);
}

// ---------------------------------------------------------------- host-side launcher
extern "C" void kernel_launch(void* const* d_in, const int* in_sizes, int n_in,
                              void* d_out, int out_size, void* d_ws, size_t ws_size,
                              hipStream_t stream) {
  const float*      F     = (const float*)d_in[0];       // [N,128] f32
  const long long*  edges = (const long long*)d_in[1];   // [2,E]   i64
  const float*      W     = (const float*)d_in[2];       // [128,128] f32
  const float*      bias  = (const float*)d_in[3];       // [128]   f32
  float*            out   = (float*)d_out;               // [N,128] f32

  int       n_nodes = in_sizes[0] / NFEAT;
  long long nE      = (long long)in_sizes[1] / 2;
  const long long* rowp = edges;
  const long long* colp = edges + nE;

  // workspace: [deg/dis: N floats][pad to 256][X: N*128 floats]
  unsigned* deg  = (unsigned*)d_ws;
  size_t    xoff = (((size_t)n_nodes * sizeof(float)) + 255) & ~(size_t)255;
  float*    X    = (float*)((char*)d_ws + xoff);

  const int thr = 256;
  long long total = (long long)n_nodes * NFEAT;

  gcn_init<<<(int)((total + thr - 1) / thr), thr, 0, stream>>>(out, deg, n_nodes, total);
  gcn_degree<<<(int)((nE + thr - 1) / thr), thr, 0, stream>>>(rowp, deg, (int)nE);
  gcn_rsqrt<<<(n_nodes + thr - 1) / thr, thr, 0, stream>>>(deg, n_nodes);

  int n_strips = (n_nodes + 15) / 16;
  gcn_gemm<<<(n_strips + 7) / 8, 256, 0, stream>>>(F, W, bias, X, n_nodes, n_strips);

  long long sthreads = (nE + (long long)n_nodes) * 32;
  gcn_scatter<<<(int)((sthreads + thr - 1) / thr), thr, 0, stream>>>(
      rowp, colp, X, (const float*)deg, out, nE, n_nodes);
}